// GRUModule_66554813219420
// MI455X (gfx1250) — compile-verified
//
#include <hip/hip_runtime.h>
#include <hip/hip_bf16.h>
#include <stdint.h>

// Problem constants (match reference): T, B, D, H
#define TT 256
#define BB 512
#define DD 512
#define HH 512
#define LN_EPS 1e-5f

// K-chunking / LDS staging geometry
#define KC 64                      // K elements staged per chunk
#define NCH 16                     // 8 chunks over H (w_hh) + 8 over D (w_ih)
#define ROWSTRIDE 68               // 64 floats + 4 pad (TDM pad) -> conflict-free ds_load_b64
#define GATEF (16 * ROWSTRIDE)     // floats per 16-row gate slab
#define BUFF (3 * GATEF)           // floats per staging buffer (3 gates)
#define GATEBYTES (GATEF * 4)
#define BUFBYTES (BUFF * 4)

typedef __attribute__((ext_vector_type(2))) float v2f;
typedef __attribute__((ext_vector_type(8))) float v8f;
typedef __attribute__((ext_vector_type(4))) unsigned int v4u;
typedef __attribute__((ext_vector_type(8))) int v8i;
typedef __attribute__((ext_vector_type(4))) int v4i;

// Exact-fp32 matrix op: D = A(16x4) x B(4x16) + C(16x16)
__device__ __forceinline__ v8f wmma4(v2f a, v2f b, v8f c) {
    return __builtin_amdgcn_wmma_f32_16x16x4_f32(
        false, a, false, b, (short)0, c, false, false);
}

__device__ __forceinline__ v2f ld2(const float* p) {
    const float2 t = *(const float2*)p;
    v2f r; r[0] = t.x; r[1] = t.y;
    return r;
}

__device__ __forceinline__ float sigmoidf_fast(float x) {
    return 1.0f / (1.0f + __expf(-x));
}

// Issue one TDM 2D tile load: 16 rows x KC cols of fp32 from a row-major
// tensor (row stride `stride0` elements) into LDS at byte offset `lds_byte`.
// Pad config inserts 4 DWORDs after every 64 DWORDs -> LDS row stride = 68 floats.
__device__ __forceinline__ void tdm_load_tile16(
    const float* gsrc, unsigned lds_byte,
    unsigned tensor_w, unsigned tensor_h, unsigned stride0)
{
    const unsigned long long ga = (unsigned long long)(uintptr_t)gsrc;
    v4u g0;
    g0[0] = 1u;                                            // count=1, user mode
    g0[1] = lds_byte;                                      // lds_addr
    g0[2] = (unsigned)(ga & 0xffffffffu);                  // global_addr[31:0]
    g0[3] = (unsigned)((ga >> 32) & 0x01ffffffu)           // global_addr[56:32]
          | (2u << 30);                                    // type=2 (image)
    v8i g1;
    g1[0] = (int)((2u << 16)      // data_size = 4 bytes
                | (1u << 20)      // pad_enable
                | (5u << 22)      // pad_interval: 64 DWORDs
                | (3u << 25));    // pad_amount: 4 DWORDs
    g1[1] = (int)((tensor_w & 0xffffu) << 16);             // tensor_dim0[15:0]
    g1[2] = (int)(((tensor_w >> 16) & 0xffffu)
                | ((tensor_h & 0xffffu) << 16));           // dim0[31:16] | dim1[15:0]
    g1[3] = (int)(((tensor_h >> 16) & 0xffffu)
                | ((unsigned)KC << 16));                   // dim1[31:16] | tile_dim0
    g1[4] = 16;                                            // tile_dim1 (tile_dim2=0)
    g1[5] = (int)stride0;                                  // tensor_dim0_stride lo32
    g1[6] = 0;                                             // stride hi16 | dim1_stride lo16
    g1[7] = 0;
    const v4i z4 = {0, 0, 0, 0};
    const v8i z8 = {0, 0, 0, 0, 0, 0, 0, 0};
    // 6-arg toolchain variant: (g0, g1, g2, g3, g4, cpol)
    __builtin_amdgcn_tensor_load_to_lds(g0, g1, z4, z4, z8, 0);
}

// One recurrence step, fused with the x @ w_ih^T contribution.
// Block = 8 waves sharing one hidden tile j0 (weights TDM-staged in LDS),
// each wave owns one 16(batch) x 16(hidden) output tile.
__global__ __launch_bounds__(256) void gru_step_kernel(
    const float* __restrict__ x,      // (T,B,D)
    const float* __restrict__ masks,  // (T,B)
    const float* __restrict__ w_ih,   // (3H,D) row-major
    const float* __restrict__ w_hh,   // (3H,H) row-major
    const float* __restrict__ b_ih,   // (3H)
    const float* __restrict__ b_hh,   // (3H)
    const float* __restrict__ h_prev, // (B,H)
    float* __restrict__ h_next,       // (B,H)
    float* __restrict__ y,            // (T,B,H) pre-LN hidden states
    int t)
{
    __shared__ float smem[2 * BUFF];  // double-buffered weight slabs (~26 KB)

    const int wave = threadIdx.x >> 5;
    const int lane = threadIdx.x & 31;
    const int lo   = lane & 15;   // A: M row / B: N col
    const int hi   = lane >> 4;   // selects K half (K = 2*hi, 2*hi+1)

    const int j0 = (blockIdx.x & 31) * 16;                 // hidden tile (shared)
    const int b0 = ((blockIdx.x >> 5) * 8 + wave) * 16;    // batch tile (per wave)

    // Generic->LDS: low 32 bits of a shared pointer are the LDS byte offset.
    const unsigned lds0 = (unsigned)(uintptr_t)&smem[0];

    v8f acc_r  = {};
    v8f acc_z  = {};
    v8f acc_xn = {};
    v8f acc_hn = {};

    const float mA = masks[(size_t)t * BB + b0 + lo];
    const float* aph = h_prev + (size_t)(b0 + lo) * HH + 2 * hi;
    const float* apx = x + ((size_t)t * BB + (b0 + lo)) * DD + 2 * hi;

    auto issue = [&](int c) {
        const float* src = (c < 8) ? w_hh : w_ih;
        const int k0 = (c & 7) * KC;
        const unsigned lbase = lds0 + (unsigned)(c & 1) * BUFBYTES;
#pragma unroll
        for (int g = 0; g < 3; ++g) {
            tdm_load_tile16(src + (size_t)(g * HH + j0) * HH + k0,
                            lbase + (unsigned)g * GATEBYTES,
                            HH, 3 * HH, HH);
        }
    };

    if (wave == 0) issue(0);

    for (int c = 0; c < NCH; ++c) {
        if (wave == 0) {
            if (c + 1 < NCH) {
                issue(c + 1);                                  // prefetch next chunk
                __builtin_amdgcn_s_wait_tensorcnt((short)3);   // current chunk done
            } else {
                __builtin_amdgcn_s_wait_tensorcnt((short)0);
            }
        }
        __syncthreads();   // staged slab visible to all waves

        const float* bb = smem + (c & 1) * BUFF;
        const int boff = lo * ROWSTRIDE + 2 * hi;
        if (c < 8) {
            // gh = (h * mask) @ w_hh^T
            const float* ap = aph + c * KC;
#pragma unroll 4
            for (int s = 0; s < KC; s += 4) {
                v2f a = ld2(ap + s);
                a[0] *= mA; a[1] *= mA;
                acc_r  = wmma4(a, ld2(bb + 0 * GATEF + boff + s), acc_r);
                acc_z  = wmma4(a, ld2(bb + 1 * GATEF + boff + s), acc_z);
                acc_hn = wmma4(a, ld2(bb + 2 * GATEF + boff + s), acc_hn);
            }
        } else {
            // xp = x_t @ w_ih^T
            const float* ap = apx + (c - 8) * KC;
#pragma unroll 4
            for (int s = 0; s < KC; s += 4) {
                v2f a = ld2(ap + s);
                acc_r  = wmma4(a, ld2(bb + 0 * GATEF + boff + s), acc_r);
                acc_z  = wmma4(a, ld2(bb + 1 * GATEF + boff + s), acc_z);
                acc_xn = wmma4(a, ld2(bb + 2 * GATEF + boff + s), acc_xn);
            }
        }
        __syncthreads();   // all reads done before this buffer is refilled
    }

    // Epilogue: gates + state update.
    // C/D layout: VGPR v holds (M = v + 8*hi, N = lo).
    const int j = j0 + lo;
    const float bihr = b_ih[j],          bhhr = b_hh[j];
    const float bihz = b_ih[HH + j],     bhhz = b_hh[HH + j];
    const float bihn = b_ih[2 * HH + j], bhhn = b_hh[2 * HH + j];

#pragma unroll
    for (int v = 0; v < 8; ++v) {
        const int b = b0 + v + 8 * hi;
        const float rg = sigmoidf_fast(acc_r[v] + bihr + bhhr);
        const float zg = sigmoidf_fast(acc_z[v] + bihz + bhhz);
        const float ng = tanhf(acc_xn[v] + bihn + rg * (acc_hn[v] + bhhn));
        const float hm = h_prev[(size_t)b * HH + j] * masks[(size_t)t * BB + b];
        const float hnew = (1.0f - zg) * ng + zg * hm;
        h_next[(size_t)b * HH + j] = hnew;
        y[((size_t)t * BB + b) * HH + j] = hnew;
    }
}

// In-place LayerNorm over the last dim (H=512). One block per (t,b) row.
__global__ __launch_bounds__(256) void ln_kernel(
    float* __restrict__ y,
    const float* __restrict__ ln_w,
    const float* __restrict__ ln_b)
{
    __shared__ float s1[256];
    __shared__ float s2[256];
    float* p = y + (size_t)blockIdx.x * HH;
    const int tid = threadIdx.x;

    const float a = p[tid];
    const float b = p[tid + 256];
    s1[tid] = a + b;
    s2[tid] = a * a + b * b;
    __syncthreads();
#pragma unroll
    for (int off = 128; off > 0; off >>= 1) {
        if (tid < off) {
            s1[tid] += s1[tid + off];
            s2[tid] += s2[tid + off];
        }
        __syncthreads();
    }
    const float mu  = s1[0] * (1.0f / HH);
    const float var = s2[0] * (1.0f / HH) - mu * mu;
    const float inv = rsqrtf(var + LN_EPS);
    p[tid]       = (a - mu) * inv * ln_w[tid]       + ln_b[tid];
    p[tid + 256] = (b - mu) * inv * ln_w[tid + 256] + ln_b[tid + 256];
}

__global__ void copy_kernel(const float* __restrict__ src,
                            float* __restrict__ dst, int n)
{
    const int i = blockIdx.x * blockDim.x + threadIdx.x;
    if (i < n) dst[i] = src[i];
}

extern "C" void kernel_launch(void* const* d_in, const int* in_sizes, int n_in,
                              void* d_out, int out_size, void* d_ws, size_t ws_size,
                              hipStream_t stream) {
    // setup_inputs order: x, rnn_states, masks, w_ih, w_hh, b_ih, b_hh, ln_w, ln_b
    const float* x      = (const float*)d_in[0];
    const float* rnn    = (const float*)d_in[1];
    const float* masks  = (const float*)d_in[2];
    const float* w_ih   = (const float*)d_in[3];
    const float* w_hh   = (const float*)d_in[4];
    const float* b_ih   = (const float*)d_in[5];
    const float* b_hh   = (const float*)d_in[6];
    const float* ln_w   = (const float*)d_in[7];
    const float* ln_b   = (const float*)d_in[8];

    float* y  = (float*)d_out;                       // (T,B,H)
    float* hT = y + (size_t)TT * BB * HH;            // (B,1,H)

    float* hbuf0 = (float*)d_ws;                     // (B,H)
    float* hbuf1 = hbuf0 + (size_t)BB * HH;          // (B,H)

    const int nH = BB * HH;

    // h0 = rnn_states (re-initialized every call; ws is not persistent)
    copy_kernel<<<(nH + 255) / 256, 256, 0, stream>>>(rnn, hbuf0, nH);

    // Sequential recurrence: one fused GEMM+gates kernel per timestep.
    for (int t = 0; t < TT; ++t) {
        const float* hp = (t & 1) ? hbuf1 : hbuf0;
        float*       hn = (t & 1) ? hbuf0 : hbuf1;
        gru_step_kernel<<<128, 256, 0, stream>>>(
            x, masks, w_ih, w_hh, b_ih, b_hh, hp, hn, y, t);
    }

    // LayerNorm in place over d_out's y region.
    ln_kernel<<<TT * BB, 256, 0, stream>>>(y, ln_w, ln_b);

    // Final hidden state (T even -> ends in hbuf0).
    copy_kernel<<<(nH + 255) / 256, 256, 0, stream>>>(hbuf0, hT, nH);
}